// FasterRCNN_16913581211798
// MI455X (gfx1250) — compile-verified
//
#include <hip/hip_runtime.h>
#include <stdint.h>

// FasterRCNN proposal->GT matcher for MI455X (gfx1250).
// B=16, N=32768 proposals, M=64 GT boxes (XYXY + class).
// Memory-bound (~18 MB @ 23.3 TB/s); GT tile (1280 B) staged in LDS via the
// CDNA5 Tensor Data Mover, consumed via LDS broadcast by all 8 waves.

#define BATCH 16
#define NPROP 32768
#define NGT 64
#define TPB 256

typedef unsigned int uint32x4 __attribute__((ext_vector_type(4)));
typedef int int32x4 __attribute__((ext_vector_type(4)));
typedef int int32x8 __attribute__((ext_vector_type(8)));

__global__ __launch_bounds__(TPB) void fasterrcnn_match_kernel(
    const float* __restrict__ proposals,   // (B, N, 4)
    const float* __restrict__ gt_boxes,    // (B, M, 5)
    float* __restrict__ out)               // (B, N, 5)
{
    __shared__ float sgt[NGT * 5];  // 1280 bytes: GT tile for this batch

    const int gid = blockIdx.x * TPB + threadIdx.x;   // global proposal id
    const int b   = blockIdx.x / (NPROP / TPB);       // uniform per block

    const float* __restrict__ gsrc = gt_boxes + (size_t)b * (NGT * 5);

#if __has_builtin(__builtin_amdgcn_tensor_load_to_lds)
    if (threadIdx.x == 0) {
        // ---- Build Tensor DMA Descriptor (D#): 1-D tile of 320 f32 ----
        const uint64_t ga  = (uint64_t)(uintptr_t)gsrc;       // global tile addr
        const uint32_t lds = (uint32_t)(uintptr_t)(&sgt[0]);  // LDS byte offset

        uint32x4 g0;
        g0[0] = 1u;                                   // count=1 (valid user D#)
        g0[1] = lds;                                  // lds_addr
        g0[2] = (uint32_t)ga;                         // global_addr[31:0]
        g0[3] = ((uint32_t)(ga >> 32) & 0x1FFFFFFu)   // global_addr[56:32]
              | (2u << 30);                           // type = 2 ("image")

        int32x8 g1;
        g1[0] = (int)(2u << 16);          // data_size = 2 -> 4-byte elements
        g1[1] = (int)(320u << 16);        // tensor_dim0[15:0]=320 (atomic_barrier=0)
        g1[2] = (int)(1u << 16);          // tensor_dim0 hi=0, tensor_dim1 lo=1
        g1[3] = (int)(320u << 16);        // tensor_dim1 hi=0, tile_dim0=320
        g1[4] = 1;                        // tile_dim1=1, tile_dim2=0
        g1[5] = 320;                      // tensor_dim0_stride lo
        g1[6] = (int)(320u << 16);        // stride0 hi=0, tensor_dim1_stride lo16=320
        g1[7] = 0;                        // tensor_dim1_stride hi
        // Benign groups 2/3 (dims = 1 so a 5-D interpretation stays in-bounds).
        int32x4 g2; g2[0] = 1; g2[1] = 1; g2[2] = 0; g2[3] = 0;
        int32x4 g3; g3[0] = 0; g3[1] = (int)(1u << 16); g3[2] = 0; g3[3] = 0;

#if defined(__clang_major__) && (__clang_major__ >= 23)
        int32x8 gx; gx[0]=0; gx[1]=0; gx[2]=0; gx[3]=0; gx[4]=0; gx[5]=0; gx[6]=0; gx[7]=0;
        __builtin_amdgcn_tensor_load_to_lds(g0, g1, g2, g3, gx, 0);
#else
        __builtin_amdgcn_tensor_load_to_lds(g0, g1, g2, g3, 0);
#endif
        __builtin_amdgcn_s_wait_tensorcnt(0);
    }
#else
    for (int i = threadIdx.x; i < NGT * 5; i += TPB) sgt[i] = gsrc[i];
#endif
    __syncthreads();

    // ---- Per-proposal IoU argmax over 64 GT boxes (division-free) ----
    const float4 p = reinterpret_cast<const float4*>(proposals)[gid];
    const float area1 = (p.z - p.x) * (p.w - p.y);

    float best_inter = -1.0f;   // encodes iou = -1 (matches invalid-GT masking)
    float best_union = 1.0f;
    int   best_j = 0;

#pragma unroll 8
    for (int j = 0; j < NGT; ++j) {
        const float gx1 = sgt[j * 5 + 0];
        const float gy1 = sgt[j * 5 + 1];
        const float gx2 = sgt[j * 5 + 2];
        const float gy2 = sgt[j * 5 + 3];
        const float gcl = sgt[j * 5 + 4];

        const float x1 = fmaxf(p.x, gx1);
        const float y1 = fmaxf(p.y, gy1);
        const float x2 = fminf(p.z, gx2);
        const float y2 = fminf(p.w, gy2);
        const float inter = fmaxf(x2 - x1, 0.0f) * fmaxf(y2 - y1, 0.0f);
        const float uni   = area1 + (gx2 - gx1) * (gy2 - gy1) - inter;

        // iou_j > iou_best  <=>  inter*best_union > best_inter*uni  (uni > 0)
        const bool better = (gcl != -1.0f) && (inter * best_union > best_inter * uni);
        if (better) { best_inter = inter; best_union = uni; best_j = j; }
    }

    // Thresholds: quality <= 0.5 -> background(-1); (0.5, 0.6) -> neutral(-1e8).
    const bool bg      = (best_inter <= 0.5f * best_union);
    const bool neutral = (!bg) && (best_inter < 0.6f * best_union);

    float* o = out + (size_t)gid * 5;
#pragma unroll
    for (int k = 0; k < 5; ++k) {
        float v = sgt[best_j * 5 + k];
        v = neutral ? -1.0e8f : v;
        v = bg ? -1.0f : v;
        o[k] = v;
    }
}

extern "C" void kernel_launch(void* const* d_in, const int* in_sizes, int n_in,
                              void* d_out, int out_size, void* d_ws, size_t ws_size,
                              hipStream_t stream) {
    (void)in_sizes; (void)n_in; (void)out_size; (void)d_ws; (void)ws_size;
    const float* proposals = (const float*)d_in[0];  // (16, 32768, 4) f32
    const float* gt_boxes  = (const float*)d_in[1];  // (16, 64, 5)    f32
    float* out             = (float*)d_out;          // (16, 32768, 5) f32

    const int blocks = (BATCH * NPROP) / TPB;        // 2048 blocks x 256 threads
    fasterrcnn_match_kernel<<<blocks, TPB, 0, stream>>>(proposals, gt_boxes, out);
}